// RGCNLayer_84516366450864
// MI455X (gfx1250) — compile-verified
//
#include <hip/hip_runtime.h>
#include <hip/hip_bf16.h>

// RGCN layer, simplified per the reference's last-edge-wins semantics:
//   agg[d] = h[src[e*]] @ W[rel[e*]]  for e* = argmax edge index with dst==d
//   out    = h @ W_self + agg
#define N_NODES 50000
#define N_EDGES 100000
#define IN_DIM  64
#define OUT_DIM 64
#define ROW_TILES (N_NODES / 16)   // 3125, exact

typedef __attribute__((ext_vector_type(2))) float v2f;
typedef __attribute__((ext_vector_type(8))) float v8f;

// ---------------------------------------------------------------- kernel 1
__global__ __launch_bounds__(256) void k_init_lastidx(int* __restrict__ last_idx) {
    int i = blockIdx.x * 256 + threadIdx.x;
    if (i < N_NODES) last_idx[i] = -1;
}

// ---------------------------------------------------------------- kernel 2
// last_idx[d] = max edge index with dst==d  (replaces segment_max; deterministic)
__global__ __launch_bounds__(256) void k_lastidx(const int* __restrict__ edges,
                                                 int* __restrict__ last_idx) {
    int e = blockIdx.x * 256 + threadIdx.x;
    if (e < N_EDGES) {
        int d = edges[3 * e + 2];
        atomicMax(&last_idx[d], e);
    }
}

// ---------------------------------------------------------------- kernel 3
// out = h @ W_self via V_WMMA_F32_16X16X4_F32.
// One wave32 per 16-row x 64-col output tile; 8 waves (8 tiles) per block.
// LDS: W_self staged TRANSPOSED (sWt[n][k], padded stride) so each B fragment
// is one contiguous 8-byte ds_load_b64 into an aligned VGPR pair; per-wave
// 16x64 A tile with the same padded stride for conflict-free frag reads.
#define PAD_STRIDE 68   // even (8B-aligned v2f) ; bank = (4*row + k) % 64
__global__ __launch_bounds__(256) void k_selfloop_wmma(const float* __restrict__ h,
                                                       const float* __restrict__ wself,
                                                       float* __restrict__ out) {
    __shared__ float sWt[64 * PAD_STRIDE];   // sWt[n*68 + k] = W_self[k][n]
    __shared__ float sA[8][16 * PAD_STRIDE]; // per-wave A tile, padded rows

    const int tid  = threadIdx.x;
    const int wave = tid >> 5;
    const int lane = tid & 31;
    const int tile = blockIdx.x * 8 + wave;

    // Stage W_self transposed (coalesced global read, scattered LDS write)
    for (int i = tid; i < 64 * 64; i += 256) {
        int k = i >> 6, n = i & 63;
        sWt[n * PAD_STRIDE + k] = wself[i];
    }

    // Stage this wave's 16x64 A tile (clamp rows so idle waves stay in-bounds)
    {
        int t = (tile < ROW_TILES) ? tile : (ROW_TILES - 1);
        for (int r = 0; r < 16; ++r) {
            int row = t * 16 + r;
            v2f a = *(const v2f*)(h + row * IN_DIM + 2 * lane);
            *(v2f*)(&sA[wave][r * PAD_STRIDE + 2 * lane]) = a;
        }
    }
    __syncthreads();

    if (tile < ROW_TILES) {
        const int half = lane >> 4;      // 0: holds K=k0,k0+1   1: K=k0+2,k0+3
        const int mn   = lane & 15;      // A: row M ; B/C: column N (within tile)

        v8f acc[4];
        #pragma unroll
        for (int nt = 0; nt < 4; ++nt) acc[nt] = (v8f){0.f,0.f,0.f,0.f,0.f,0.f,0.f,0.f};

        #pragma unroll
        for (int kt = 0; kt < 16; ++kt) {           // K = 64 in steps of 4
            const int k0 = kt * 4 + 2 * half;
            // A frag (16x4 fp32): lane holds A[mn][k0], A[mn][k0+1]
            v2f a = *(const v2f*)(&sA[wave][mn * PAD_STRIDE + k0]);
            #pragma unroll
            for (int nt = 0; nt < 4; ++nt) {
                // B frag (4x16 fp32): lane holds W[k0][n], W[k0+1][n], n=16*nt+mn
                v2f b = *(const v2f*)(&sWt[(nt * 16 + mn) * PAD_STRIDE + k0]);
                acc[nt] = __builtin_amdgcn_wmma_f32_16x16x4_f32(
                    /*neg_a=*/false, a, /*neg_b=*/false, b,
                    /*c_mod=*/(short)0, acc[nt],
                    /*reuse_a=*/false, /*reuse_b=*/false);
            }
        }

        // C/D layout: VGPR j, lanes 0-15 -> M=j, lanes 16-31 -> M=j+8; N=mn
        #pragma unroll
        for (int nt = 0; nt < 4; ++nt) {
            #pragma unroll
            for (int j = 0; j < 8; ++j) {
                int row = tile * 16 + j + 8 * half;
                out[row * OUT_DIM + nt * 16 + mn] = acc[nt][j];
            }
        }
    }
}

// ---------------------------------------------------------------- kernel 4
// out[d][:] += h[src[e*]] @ W[rel[e*]] for nodes with an incoming edge.
// 64 threads per node (one output column each); W columns coalesced.
__global__ __launch_bounds__(256) void k_msg_add(const float* __restrict__ h,
                                                 const int* __restrict__ edges,
                                                 const float* __restrict__ weight,
                                                 const int* __restrict__ last_idx,
                                                 float* __restrict__ out) {
    int node = blockIdx.x * 4 + (threadIdx.x >> 6);
    int o    = threadIdx.x & 63;
    if (node >= N_NODES) return;
    int e = last_idx[node];
    if (e < 0) return;                      // count==0 -> agg is 0

    int s = edges[3 * e + 0];
    int r = edges[3 * e + 1];
    const float* __restrict__ hs = h + s * IN_DIM;
    const float* __restrict__ w  = weight + r * (IN_DIM * OUT_DIM);
    __builtin_prefetch(w, 0, 1);            // global_prefetch_b8 (L2-resident W)

    float acc = 0.f;
    #pragma unroll 8
    for (int i = 0; i < IN_DIM; ++i)
        acc = fmaf(hs[i], w[i * OUT_DIM + o], acc);

    out[node * OUT_DIM + o] += acc;
}

// ---------------------------------------------------------------- launch
extern "C" void kernel_launch(void* const* d_in, const int* in_sizes, int n_in,
                              void* d_out, int out_size, void* d_ws, size_t ws_size,
                              hipStream_t stream) {
    const float* input_h = (const float*)d_in[0];   // (50000, 64) f32
    const int*   edges   = (const int*)  d_in[1];   // (100000, 3) i32 [src,rel,dst]
    const float* weight  = (const float*)d_in[2];   // (200, 64, 64) f32
    const float* wself   = (const float*)d_in[3];   // (64, 64) f32
    float*       out     = (float*)d_out;           // (50000, 64) f32
    int*         last_idx = (int*)d_ws;             // 50000 ints of scratch

    k_init_lastidx<<<(N_NODES + 255) / 256, 256, 0, stream>>>(last_idx);
    k_lastidx<<<(N_EDGES + 255) / 256, 256, 0, stream>>>(edges, last_idx);
    k_selfloop_wmma<<<(ROW_TILES + 7) / 8, 256, 0, stream>>>(input_h, wself, out);
    k_msg_add<<<(N_NODES + 3) / 4, 256, 0, stream>>>(input_h, edges, weight,
                                                     last_idx, out);
}